// KeypointBatchToPoseGT_80831284510871
// MI455X (gfx1250) — compile-verified
//
#include <hip/hip_runtime.h>

// KeypointBatchToPoseGT — pure streaming elementwise transform.
// HBM-bound: ~107 MB in + ~214 MB out -> ~14us floor at 23.3 TB/s.
// Strategy: 4 triples/thread => all global traffic is aligned B128 with TH_NT.
// K is a compile-time constant so i/K folds to a magic mulhi (no runtime rcp).

typedef float f32x4 __attribute__((ext_vector_type(4)));
typedef int   i32x4 __attribute__((ext_vector_type(4)));

#define DEV_INLINE __device__ __forceinline__

// Constants (match reference float32 semantics exactly)
constexpr unsigned kK       = 17u;      // keypoints per batch (fixed by reference)
constexpr float kInvDeltaXY = 0.125f;   // exact reciprocal of 8.0
constexpr float kMinXY      = 0.0f;
constexpr float kMaxXY      = 504.0f;   // (64-1)*8.0
constexpr float kMinZ       = 0.0f;
constexpr float kMaxZ       = 3.15f;    // f32(63*0.05) == 0x4049999A, same as JAX clip bound

DEV_INLINE float clampf(float v, float lo, float hi) {
  return fminf(fmaxf(v, lo), hi);
}

__global__ __launch_bounds__(256)
void keypoint_gt_kernel(const float* __restrict__ x,      // [N,3]
                        float*       __restrict__ out_xy, // [N,2]
                        float*       __restrict__ out_z,  // [N]
                        int*         __restrict__ out_idx,// [N,3]
                        int n) {
  const int gid = blockIdx.x * blockDim.x + threadIdx.x;
  const int i0  = gid * 4;
  if (i0 >= n) return;

  if (i0 + 4 <= n) {
    // ---- fast path: 4 triples, all-B128 traffic ----
    const f32x4* xp = (const f32x4*)(x + (size_t)i0 * 3);
    const f32x4 a = __builtin_nontemporal_load(xp + 0);  // x0 y0 z0 x1
    const f32x4 b = __builtin_nontemporal_load(xp + 1);  // y1 z1 x2 y2
    const f32x4 c = __builtin_nontemporal_load(xp + 2);  // z2 x3 y3 z3

    const float xs[4] = {a.x, a.w, b.z, c.y};
    const float ys[4] = {a.y, b.x, b.w, c.z};
    const float zs[4] = {a.z, b.y, c.x, c.w};

    float cx[4], cy[4], cz[4];
    int   bi[4], ix[4], iy[4];
#pragma unroll
    for (int j = 0; j < 4; ++j) {
      cx[j] = clampf(xs[j], kMinXY, kMaxXY);
      cy[j] = clampf(ys[j], kMinXY, kMaxXY);
      cz[j] = clampf(zs[j], kMinZ,  kMaxZ);
      ix[j] = (int)(cx[j] * kInvDeltaXY);      // truncation == astype(int32), vals >= 0
      iy[j] = (int)(cy[j] * kInvDeltaXY);
      bi[j] = (int)(((unsigned)i0 + (unsigned)j) / kK);  // compile-time magic mulhi
    }

    // gt_xy: 8 floats at 2*i0 (16B aligned)
    f32x4* oxy = (f32x4*)(out_xy + (size_t)i0 * 2);
    const f32x4 xyA = {cx[0], cy[0], cx[1], cy[1]};
    const f32x4 xyB = {cx[2], cy[2], cx[3], cy[3]};
    __builtin_nontemporal_store(xyA, oxy + 0);
    __builtin_nontemporal_store(xyB, oxy + 1);

    // gt_loc_z: 4 floats at i0 (16B aligned)
    const f32x4 zv = {cz[0], cz[1], cz[2], cz[3]};
    __builtin_nontemporal_store(zv, (f32x4*)(out_z + i0));

    // gt_index_z: 12 ints at 3*i0 (16B aligned)
    i32x4* oid = (i32x4*)(out_idx + (size_t)i0 * 3);
    const i32x4 t0 = {bi[0], ix[0], iy[0], bi[1]};
    const i32x4 t1 = {ix[1], iy[1], bi[2], ix[2]};
    const i32x4 t2 = {iy[2], bi[3], ix[3], iy[3]};
    __builtin_nontemporal_store(t0, oid + 0);
    __builtin_nontemporal_store(t1, oid + 1);
    __builtin_nontemporal_store(t2, oid + 2);
  } else {
    // ---- scalar tail (N % 4 == 0 for this workload -> statically unused) ----
    for (int i = i0; i < n; ++i) {
      const float vx = x[(size_t)i * 3 + 0];
      const float vy = x[(size_t)i * 3 + 1];
      const float vz = x[(size_t)i * 3 + 2];
      const float cxs = clampf(vx, kMinXY, kMaxXY);
      const float cys = clampf(vy, kMinXY, kMaxXY);
      out_xy[(size_t)i * 2 + 0] = cxs;
      out_xy[(size_t)i * 2 + 1] = cys;
      out_z[i] = clampf(vz, kMinZ, kMaxZ);
      out_idx[(size_t)i * 3 + 0] = (int)((unsigned)i / kK);
      out_idx[(size_t)i * 3 + 1] = (int)(cxs * kInvDeltaXY);
      out_idx[(size_t)i * 3 + 2] = (int)(cys * kInvDeltaXY);
    }
  }
}

extern "C" void kernel_launch(void* const* d_in, const int* in_sizes, int n_in,
                              void* d_out, int out_size, void* d_ws, size_t ws_size,
                              hipStream_t stream) {
  (void)n_in; (void)out_size; (void)d_ws; (void)ws_size;

  const float* x = (const float*)d_in[0];
  const int N = in_sizes[0] / 3;        // B*K = 8,912,896

  // d_out layout (return order, all 4-byte elements):
  //   [0, 2N)        gt_xy       float32
  //   [2N, 3N)       gt_loc_z    float32
  //   [3N, 6N)       gt_index_z  int32
  float* out_xy  = (float*)d_out;
  float* out_z   = out_xy + (size_t)N * 2;
  int*   out_idx = (int*)(out_xy + (size_t)N * 3);

  const int threads = 256;
  const int work    = (N + 3) / 4;                 // 4 triples per thread
  const int blocks  = (work + threads - 1) / threads;

  keypoint_gt_kernel<<<blocks, threads, 0, stream>>>(x, out_xy, out_z, out_idx, N);
}